// GraphSAGELayer_32727650795829
// MI455X (gfx1250) — compile-verified
//
#include <hip/hip_runtime.h>

typedef __attribute__((ext_vector_type(2))) float v2f;
typedef __attribute__((ext_vector_type(8))) float v8f;

#define IN_F   128
#define OUT_F  256
#define TILE_M 16
#define LDA    130   // padded LDS row stride for 128-wide tiles (even, conflict-spreading)
#define LDT    258   // padded LDS row stride for 256-wide t tile

// ---------------- Kernel 1: zero the workspace accumulators ----------------
__global__ void k_zero(float* __restrict__ p, int n) {
    int i = blockIdx.x * blockDim.x + threadIdx.x;
    if (i < n) p[i] = 0.0f;
}

// ---------------- Kernel 2: edge aggregation (gather + atomic scatter-add) ----
// One wave32 per edge. Lane l loads float4 l of the 128-float source row
// (32 lanes * 16B = whole row) and atomically adds into summed[dst].
__global__ __launch_bounds__(256)
void k_agg(const float* __restrict__ feat, const int* __restrict__ src,
           const int* __restrict__ dst, float* __restrict__ summed,
           float* __restrict__ deg, int E) {
    int wave = (blockIdx.x * blockDim.x + threadIdx.x) >> 5;  // uniform per wave
    int lane = threadIdx.x & 31;
    if (wave >= E) return;
    int s = src[wave];
    int d = dst[wave];
    const float4* row = (const float4*)(feat + (size_t)s * IN_F);
    float4 v = row[lane];
    float* o = summed + (size_t)d * IN_F + lane * 4;
    atomicAdd(o + 0, v.x);
    atomicAdd(o + 1, v.y);
    atomicAdd(o + 2, v.z);
    atomicAdd(o + 3, v.w);
    if (lane == 0) atomicAdd(deg + d, 1.0f);
}

// ---------------- Kernel 3: fused dense part, f32 WMMA ----------------------
// out = feat@W_self + relu(hn@W1 + b1)@W2 + (b_self + b2), hn = summed/max(deg,1)
// Block = 256 threads (8 waves) handles a 16-node row tile.
// Wave w owns output columns [w*32, w*32+32) as two 16x16 WMMA tiles.
__global__ __launch_bounds__(256)
void k_gemm(const float* __restrict__ feat,  const float* __restrict__ W_self,
            const float* __restrict__ b_self,const float* __restrict__ W1,
            const float* __restrict__ b1,    const float* __restrict__ W2,
            const float* __restrict__ b2,    const float* __restrict__ summed,
            const float* __restrict__ deg,   float* __restrict__ out, int n) {
    __shared__ float s_feat[TILE_M * LDA];
    __shared__ float s_hn  [TILE_M * LDA];
    __shared__ float s_t   [TILE_M * LDT];

    const int node0 = blockIdx.x * TILE_M;
    const int tid   = threadIdx.x;
    const int lane  = tid & 31;
    const int wv    = tid >> 5;

    // Stage feat tile and normalized-neighbor tile into LDS (2048 f32 each).
    for (int i = tid; i < TILE_M * IN_F; i += 256) {
        int r = i >> 7;            // 0..15
        int c = i & (IN_F - 1);    // 0..127
        int rr = node0 + r; if (rr > n - 1) rr = n - 1;
        float dg  = deg[rr];
        float inv = 1.0f / fmaxf(dg, 1.0f);
        s_feat[r * LDA + c] = feat  [(size_t)rr * IN_F + c];
        s_hn  [r * LDA + c] = summed[(size_t)rr * IN_F + c] * inv;
    }
    __syncthreads();

    // ISA 7.12.2 fragment mappings (wave32, 32-bit data):
    // A 16x4 : lane holds (M = lane&15, K = k0 + 2*(lane>>4) + {0,1})  -> v2f
    // B 4x16 : lane holds (N = lane&15, K = k0 + 2*(lane>>4) + {0,1})  -> v2f
    // C 16x16: VGPR i -> (M = i + 8*(lane>>4), N = lane&15)
    const int arow = lane & 15;
    const int koff = (lane >> 4) * 2;
    const int bcol = lane & 15;
    const int crow = (lane >> 4) * 8;

    // ---- Phase 1: t = relu(hn @ W1 + b1) for this wave's 32 columns ----
    for (int t2 = 0; t2 < 2; ++t2) {
        const int colbase = (wv * 2 + t2) * 16;
        v8f acc = {};
        for (int k0 = 0; k0 < IN_F; k0 += 4) {
            int k = k0 + koff;
            v2f a = *(const v2f*)(&s_hn[arow * LDA + k]);
            v2f b;
            b.x = W1[(size_t)(k)     * OUT_F + colbase + bcol];
            b.y = W1[(size_t)(k + 1) * OUT_F + colbase + bcol];
            acc = __builtin_amdgcn_wmma_f32_16x16x4_f32(
                false, a, false, b, (short)0, acc, false, false);
        }
        int cc = colbase + (lane & 15);
        float bias = b1[cc];
#pragma unroll
        for (int i = 0; i < 8; ++i) {
            float v = acc[i] + bias;
            s_t[(crow + i) * LDT + cc] = fmaxf(v, 0.0f);
        }
    }
    __syncthreads();

    // ---- Phase 2: out = feat @ W_self  +  t @ W2  + (b_self + b2) ----
    for (int t2 = 0; t2 < 2; ++t2) {
        const int colbase = (wv * 2 + t2) * 16;
        v8f acc = {};
        // self path, K = 128
        for (int k0 = 0; k0 < IN_F; k0 += 4) {
            int k = k0 + koff;
            v2f a = *(const v2f*)(&s_feat[arow * LDA + k]);
            v2f b;
            b.x = W_self[(size_t)(k)     * OUT_F + colbase + bcol];
            b.y = W_self[(size_t)(k + 1) * OUT_F + colbase + bcol];
            acc = __builtin_amdgcn_wmma_f32_16x16x4_f32(
                false, a, false, b, (short)0, acc, false, false);
        }
        // neighbor MLP second layer, K = 256
        for (int k0 = 0; k0 < OUT_F; k0 += 4) {
            int k = k0 + koff;
            v2f a = *(const v2f*)(&s_t[arow * LDT + k]);
            v2f b;
            b.x = W2[(size_t)(k)     * OUT_F + colbase + bcol];
            b.y = W2[(size_t)(k + 1) * OUT_F + colbase + bcol];
            acc = __builtin_amdgcn_wmma_f32_16x16x4_f32(
                false, a, false, b, (short)0, acc, false, false);
        }
        int cc = colbase + (lane & 15);
        float bias = b_self[cc] + b2[cc];
#pragma unroll
        for (int i = 0; i < 8; ++i) {
            int r = node0 + crow + i;
            if (r < n) out[(size_t)r * OUT_F + cc] = acc[i] + bias;
        }
    }
}

extern "C" void kernel_launch(void* const* d_in, const int* in_sizes, int n_in,
                              void* d_out, int out_size, void* d_ws, size_t ws_size,
                              hipStream_t stream) {
    const float* feat   = (const float*)d_in[0];
    const float* W_self = (const float*)d_in[1];
    const float* b_self = (const float*)d_in[2];
    const float* W1     = (const float*)d_in[3];
    const float* b1     = (const float*)d_in[4];
    const float* W2     = (const float*)d_in[5];
    const float* b2     = (const float*)d_in[6];
    const int*   src    = (const int*)d_in[7];
    const int*   dst    = (const int*)d_in[8];
    float*       out    = (float*)d_out;

    const int N = in_sizes[0] / IN_F;   // 50000
    const int E = in_sizes[7];          // 800000

    float* summed = (float*)d_ws;                  // N * 128 f32
    float* deg    = summed + (size_t)N * IN_F;     // N f32

    int nz = N * IN_F + N;
    k_zero<<<(nz + 255) / 256, 256, 0, stream>>>(summed, nz);

    // one wave32 per edge -> E*32 threads
    long long tthreads = (long long)E * 32;
    int aggBlocks = (int)((tthreads + 255) / 256);
    k_agg<<<aggBlocks, 256, 0, stream>>>(feat, src, dst, summed, deg, E);

    int gBlocks = (N + TILE_M - 1) / TILE_M;       // 3125
    k_gemm<<<gBlocks, 256, 0, stream>>>(feat, W_self, b_self, W1, b1, W2, b2,
                                        summed, deg, out, N);
}